// DRM_Matching_28518582845475
// MI455X (gfx1250) — compile-verified
//
#include <hip/hip_runtime.h>
#include <math.h>

// Problem constants (fixed by the reference setup_inputs()).
constexpr int Bb = 16, Hh = 50, Ss = 128, Ll = 2, Dd = 128, Kk = 10;
constexpr float EPSV = 1e-12f;
constexpr float THRESH = -1e30f;

typedef __attribute__((ext_vector_type(2))) float v2f;
typedef __attribute__((ext_vector_type(4))) float v4f;
typedef __attribute__((ext_vector_type(8))) float v8f;

#if defined(__HIP_DEVICE_COMPILE__) && __has_builtin(__builtin_amdgcn_wmma_f32_16x16x4_f32)
#define USE_WMMA_F32X4 1
#else
#define USE_WMMA_F32X4 0
#endif

// One block per (b,h): 256 threads = 8 waves; wave t owns the 16-word tile
// [16t, 16t+16). Streams the 128KB (b,h) slice of news_embedding exactly once.
// Score dots run on V_WMMA_F32_16X16X4_F32; K-slot assignment is permuted so
// each lane feeds two WMMAs from one 16-byte global load (dot is commutative
// over K, and A/B use the same permutation).
__global__ __launch_bounds__(256) void drm_matching_kernel(
    const float* __restrict__ news,   // [B,H,S,L,D]
    const float* __restrict__ user,   // [B,1,D]
    float* __restrict__ out_terms,    // [B,H,K,L,D]
    float* __restrict__ out_ids)      // [B,H,K] (as float)
{
    __shared__ float u_s[Dd];     // normalized user vector
    __shared__ float sc[Ss];      // raw dot(mean, u_n) per word
    __shared__ float n2[Ss];      // ||mean||^2 per word
    __shared__ float topv[Kk];
    __shared__ int   topi[Kk];
    __shared__ float uinv_s;

    const int bh   = blockIdx.x;
    const int b    = bh / Hh;
    const int tid  = threadIdx.x;
    const int lane = tid & 31;
    const int wave = tid >> 5;
    const int half = lane >> 4;        // 0: lanes 0-15, 1: lanes 16-31

    // --- normalize user vector (cheap; redundant across H blocks) ---
    if (tid < Dd) u_s[tid] = user[(size_t)b * Dd + tid];
    __syncthreads();
    if (wave == 0) {
        float s = 0.f;
#pragma unroll
        for (int q = 0; q < Dd / 32; ++q) {
            float x = u_s[lane + 32 * q];
            s += x * x;
        }
#pragma unroll
        for (int o = 16; o > 0; o >>= 1) s += __shfl_xor(s, o, 32);
        if (lane == 0) uinv_s = 1.f / fmaxf(sqrtf(s), EPSV);
    }
    __syncthreads();
    if (tid < Dd) u_s[tid] *= uinv_s;
    __syncthreads();

    // --- phase 1: per-word scores via WMMA ---
    const float* E   = news + (size_t)bh * (Ss * Ll * Dd);
    const int tile   = wave * 16;
    const int word   = tile + (lane & 15);
    const float* row = E + (size_t)word * (Ll * Dd);

    v8f  c      = {};
    float dotac = 0.f;   // fallback accumulator
    float nsq   = 0.f;   // partial ||mean||^2 for this lane's d slices

#pragma unroll 2
    for (int kk = 0; kk < Dd / 8; ++kk) {
        // Permuted K assignment: this lane owns d0..d0+3; lane^16 owns the
        // other 4 of the 8-wide chunk. Two WMMAs consume the 4 values.
        const int d0 = 8 * kk + 4 * half;
        v4f e0 = *(const v4f*)(row + d0);           // level 0, 16B
        v4f e1 = *(const v4f*)(row + Dd + d0);      // level 1, 16B
        float m0 = 0.5f * (e0.x + e1.x);
        float m1 = 0.5f * (e0.y + e1.y);
        float m2 = 0.5f * (e0.z + e1.z);
        float m3 = 0.5f * (e0.w + e1.w);
        nsq += m0 * m0 + m1 * m1 + m2 * m2 + m3 * m3;
        const v4f u4 = *(const v4f*)(u_s + d0);     // ds_load_b128
#if USE_WMMA_F32X4
        v2f a0;  a0.x = m0;   a0.y = m1;
        v2f b0;  b0.x = u4.x; b0.y = u4.y;
        c = __builtin_amdgcn_wmma_f32_16x16x4_f32(
                false, a0, false, b0, (short)0, c, false, false);
        v2f a1;  a1.x = m2;   a1.y = m3;
        v2f b1;  b1.x = u4.z; b1.y = u4.w;
        c = __builtin_amdgcn_wmma_f32_16x16x4_f32(
                false, a1, false, b1, (short)0, c, false, false);
#else
        dotac += m0 * u4.x + m1 * u4.y + m2 * u4.z + m3 * u4.w;
#endif
    }

    // ||mean||^2: lane and lane^16 together cover all of D
    float nfull = nsq + __shfl_xor(nsq, 16, 32);
    if (lane < 16) n2[tile + lane] = nfull;

#if USE_WMMA_F32X4
    // D[m,n] is identical over n; lane half h holds scores for words v+8h.
    if ((lane & 15) == 0) {
#pragma unroll
        for (int v = 0; v < 8; ++v) sc[tile + 8 * half + v] = c[v];
    }
#else
    float dfull = dotac + __shfl_xor(dotac, 16, 32);
    if (lane < 16) sc[tile + lane] = dfull;
#endif
    __syncthreads();

    // --- phase 2: top-10 over S=128 (wave 0, iterative argmax, stable ties) ---
    if (wave == 0) {
        float sv[4];
#pragma unroll
        for (int q = 0; q < 4; ++q) {
            int j = lane + 32 * q;
            sv[q] = sc[j] / fmaxf(sqrtf(n2[j]), EPSV);
        }
        for (int it = 0; it < Kk; ++it) {
            float mv = -3.4e38f;
            int   mi = Ss;
#pragma unroll
            for (int q = 0; q < 4; ++q) {
                int j = lane + 32 * q;
                if (sv[q] > mv || (sv[q] == mv && j < mi)) { mv = sv[q]; mi = j; }
            }
#pragma unroll
            for (int o = 16; o > 0; o >>= 1) {
                float ov = __shfl_xor(mv, o, 32);
                int   oi = __shfl_xor(mi, o, 32);
                if (ov > mv || (ov == mv && oi < mi)) { mv = ov; mi = oi; }
            }
            if (lane == 0) { topv[it] = mv; topi[it] = mi; }
#pragma unroll
            for (int q = 0; q < 4; ++q)
                if (lane + 32 * q == mi) sv[q] = -3.4e38f;
        }
    }
    __syncthreads();

    // --- phase 3: gather + weight (coalesced; rows are L2-hot) ---
#pragma unroll
    for (int k = 0; k < Kk; ++k) {
        int   sid = topi[k];
        float wv  = topv[k];
        if (wv < THRESH) wv = 0.f;
        out_terms[((size_t)bh * Kk + k) * (Ll * Dd) + tid] =
            E[(size_t)sid * (Ll * Dd) + tid] * wv;
    }
    if (tid < Kk) out_ids[(size_t)bh * Kk + tid] = (float)topi[tid];
}

extern "C" void kernel_launch(void* const* d_in, const int* in_sizes, int n_in,
                              void* d_out, int out_size, void* d_ws, size_t ws_size,
                              hipStream_t stream) {
    (void)in_sizes; (void)n_in; (void)out_size; (void)d_ws; (void)ws_size;
    const float* news = (const float*)d_in[0];
    const float* user = (const float*)d_in[1];
    float* out_terms = (float*)d_out;
    float* out_ids   = out_terms + (size_t)Bb * Hh * Kk * Ll * Dd;

    drm_matching_kernel<<<dim3(Bb * Hh), dim3(256), 0, stream>>>(
        news, user, out_terms, out_ids);
}